// ContrastLayer_25409026523345
// MI455X (gfx1250) — compile-verified
//
#include <hip/hip_runtime.h>
#include <hip/hip_bf16.h>
#include <math.h>

typedef __attribute__((ext_vector_type(2))) float v2f;
typedef __attribute__((ext_vector_type(8))) float v8f;

#define NPOINT 40000
#define NAGENT 40000
#define NEDGE  400000
#define DMODEL 256
#define NHEAD  8
#define DHEAD  32
#define DFFN   2048

// ---------------------------------------------------------------------------
// fp32 WMMA GEMM:  C[M,N] = A[M,K] @ B (+bias, optional relu)
//   opB==0 : B is K x N row-major      (x @ W)
//   opB==1 : B is N x K row-major      (x @ W^T, torch-style weights)
// Block tile 128x64, 8 waves (4x2), each wave 32x32 via 2x2 WMMA 16x16x4 f32.
// ---------------------------------------------------------------------------
#define TM 128
#define TN 64
#define KC 32
#define ASTR 36
#define BSTR 36

__global__ __launch_bounds__(256)
void gemm_wmma_f32(const float* __restrict__ A, const float* __restrict__ B,
                   float* __restrict__ C, const float* __restrict__ bias,
                   int M, int N, int K, int opB, int relu)
{
    __shared__ __align__(16) float As[TM * ASTR];   // [m][k], k contiguous
    __shared__ __align__(16) float Bs[TN * BSTR];   // [n][k], k contiguous (B^T tile)

    const int t    = threadIdx.x;
    const int m0   = blockIdx.y * TM;
    const int n0   = blockIdx.x * TN;
    const int wid  = t >> 5;
    const int lane = t & 31;
    const int wy   = wid >> 1;          // 0..3 -> M offset wy*32
    const int wx   = wid & 1;           // 0..1 -> N offset wx*32
    const int half = lane >> 4;         // K sub-offset selector
    const int lm   = lane & 15;

    v8f acc[2][2];
    const v8f vzero = {0.f,0.f,0.f,0.f,0.f,0.f,0.f,0.f};
    acc[0][0] = vzero; acc[0][1] = vzero; acc[1][0] = vzero; acc[1][1] = vzero;

    for (int k0 = 0; k0 < K; k0 += KC) {
        // ---- stage A tile: 128x32 floats, 256 thr x 4 float4
        #pragma unroll
        for (int i = 0; i < 4; i++) {
            int idx  = t + i * 256;          // 0..1023 float4 slots
            int row  = idx >> 3;             // 8 float4 per 32-float row
            int col4 = (idx & 7) << 2;
            int gr   = m0 + row;
            float4 v = make_float4(0.f, 0.f, 0.f, 0.f);
            if (gr < M) {
                const float* src = A + (size_t)gr * K + k0 + col4;
                v = *(const float4*)src;
                if (k0 + KC < K) __builtin_prefetch(src + KC, 0, 3);
            }
            *(float4*)(As + row * ASTR + col4) = v;
        }
        // ---- stage B tile transposed into Bs[n][kk]
        if (opB) {
            #pragma unroll
            for (int i = 0; i < 2; i++) {
                int idx  = t + i * 256;      // 512 float4 slots (64x32)
                int row  = idx >> 3;
                int col4 = (idx & 7) << 2;
                *(float4*)(Bs + row * BSTR + col4) =
                    *(const float4*)(B + (size_t)(n0 + row) * K + k0 + col4);
            }
        } else {
            #pragma unroll
            for (int i = 0; i < 8; i++) {
                int idx = t + i * 256;       // 2048 scalars
                int kk  = idx >> 6;
                int n   = idx & 63;
                Bs[n * BSTR + kk] = B[(size_t)(k0 + kk) * N + n0 + n];
            }
        }
        __syncthreads();

        // ---- 8 K-steps of 4, 4 WMMAs each
        #pragma unroll
        for (int kk = 0; kk < KC; kk += 4) {
            v2f a0 = *(const v2f*)(As + (wy * 32 +  0 + lm) * ASTR + kk + 2 * half);
            v2f a1 = *(const v2f*)(As + (wy * 32 + 16 + lm) * ASTR + kk + 2 * half);
            v2f b0 = *(const v2f*)(Bs + (wx * 32 +  0 + lm) * BSTR + kk + 2 * half);
            v2f b1 = *(const v2f*)(Bs + (wx * 32 + 16 + lm) * BSTR + kk + 2 * half);
            acc[0][0] = __builtin_amdgcn_wmma_f32_16x16x4_f32(false, a0, false, b0, (short)0, acc[0][0], false, false);
            acc[0][1] = __builtin_amdgcn_wmma_f32_16x16x4_f32(false, a0, false, b1, (short)0, acc[0][1], false, false);
            acc[1][0] = __builtin_amdgcn_wmma_f32_16x16x4_f32(false, a1, false, b0, (short)0, acc[1][0], false, false);
            acc[1][1] = __builtin_amdgcn_wmma_f32_16x16x4_f32(false, a1, false, b1, (short)0, acc[1][1], false, false);
        }
        __syncthreads();
    }

    // ---- epilogue: C layout lane l, vgpr v -> row = base + v + 8*(l/16), col = base + l%16
    #pragma unroll
    for (int mi = 0; mi < 2; mi++) {
        #pragma unroll
        for (int ni = 0; ni < 2; ni++) {
            int n  = n0 + wx * 32 + ni * 16 + lm;
            float bv = bias ? bias[n] : 0.f;
            #pragma unroll
            for (int v = 0; v < 8; v++) {
                int m = m0 + wy * 32 + mi * 16 + v + 8 * half;
                if (m < M) {
                    float val = acc[mi][ni][v] + bv;
                    if (relu) val = fmaxf(val, 0.f);
                    C[(size_t)m * N + n] = val;
                }
            }
        }
    }
}

// ---------------------------------------------------------------------------
// helpers
// ---------------------------------------------------------------------------
__device__ __forceinline__ float wave_sum32(float v) {
    #pragma unroll
    for (int off = 16; off; off >>= 1) v += __shfl_xor(v, off, 32);
    return v;
}

__device__ __forceinline__ void atomicMaxF(float* addr, float val) {
    if (val >= 0.f) atomicMax((int*)addr, __float_as_int(val));
    else            atomicMin((unsigned int*)addr, __float_as_uint(val));
}

__global__ void fill_kernel(float* __restrict__ p, float v, size_t n) {
    size_t i = (size_t)blockIdx.x * blockDim.x + threadIdx.x;
    size_t stride = (size_t)gridDim.x * blockDim.x;
    for (; i < n; i += stride) p[i] = v;
}

__global__ void clamp_finite_kernel(float* __restrict__ m, size_t n) {
    size_t i = (size_t)blockIdx.x * blockDim.x + threadIdx.x;
    size_t stride = (size_t)gridDim.x * blockDim.x;
    for (; i < n; i += stride) { float v = m[i]; if (!isfinite(v)) m[i] = 0.f; }
}

// el[n,h] = sum_d H[n,h*32+d]*al[h*32+d]; er likewise. One block (256 thr) per row.
__global__ void elr_kernel(const float* __restrict__ Hf, const float* __restrict__ al,
                           const float* __restrict__ ar, float* __restrict__ el,
                           float* __restrict__ er, int nrows)
{
    int n = blockIdx.x; if (n >= nrows) return;
    int t = threadIdx.x;
    float hv = Hf[(size_t)n * DMODEL + t];
    float pl = wave_sum32(hv * al[t]);
    float pr = wave_sum32(hv * ar[t]);
    if ((t & 31) == 0) { int h = t >> 5; el[n * NHEAD + h] = pl; er[n * NHEAD + h] = pr; }
}

// pass 1: segment max of leakyrelu(el[src]+er[dst])
__global__ void edge_max_kernel(const int* __restrict__ src, const int* __restrict__ dst,
                                int dst_off, const float* __restrict__ el,
                                const float* __restrict__ er, float* __restrict__ m, int nE)
{
    int i = blockIdx.x * 256 + threadIdx.x;
    if (i >= nE * NHEAD) return;
    int e = i >> 3, h = i & 7;
    int s = src[e], d = dst[e] + dst_off;
    float v = el[s * NHEAD + h] + er[d * NHEAD + h];
    v = v > 0.f ? v : 0.2f * v;
    atomicMaxF(&m[d * NHEAD + h], v);
}

__global__ void loop_max_kernel(int base, int n, const float* __restrict__ el,
                                const float* __restrict__ er, float* __restrict__ m)
{
    int i = blockIdx.x * 256 + threadIdx.x;
    if (i >= n * NHEAD) return;
    int node = base + (i >> 3), h = i & 7;
    float v = el[node * NHEAD + h] + er[node * NHEAD + h];
    v = v > 0.f ? v : 0.2f * v;
    atomicMaxF(&m[node * NHEAD + h], v);
}

// pass 2: a = exp(e - m[dst]); s[dst,h] += a; out[dst-out_off, :] += a*H[src,:]
__global__ void edge_acc_kernel(const int* __restrict__ src, const int* __restrict__ dst,
                                int dst_off, int out_off,
                                const float* __restrict__ el, const float* __restrict__ er,
                                const float* __restrict__ m, float* __restrict__ s,
                                float* __restrict__ out, const float* __restrict__ Hsrc, int nE)
{
    int e = blockIdx.x; if (e >= nE) return;
    int t = threadIdx.x, h = t >> 5;
    int sn = src[e], dn = dst[e] + dst_off;
    float v = el[sn * NHEAD + h] + er[dn * NHEAD + h];
    v = v > 0.f ? v : 0.2f * v;
    float a = expf(v - m[dn * NHEAD + h]);
    if ((t & 31) == 0) atomicAdd(&s[dn * NHEAD + h], a);
    atomicAdd(&out[(size_t)(dn - out_off) * DMODEL + t], a * Hsrc[(size_t)sn * DMODEL + t]);
}

__global__ void loop_acc_kernel(int base, int out_off,
                                const float* __restrict__ el, const float* __restrict__ er,
                                const float* __restrict__ m, float* __restrict__ s,
                                float* __restrict__ out, const float* __restrict__ Hsrc, int n)
{
    int e = blockIdx.x; if (e >= n) return;
    int node = base + e;
    int t = threadIdx.x, h = t >> 5;
    float v = el[node * NHEAD + h] + er[node * NHEAD + h];
    v = v > 0.f ? v : 0.2f * v;
    float a = expf(v - m[node * NHEAD + h]);
    if ((t & 31) == 0) atomicAdd(&s[node * NHEAD + h], a);
    atomicAdd(&out[(size_t)(node - out_off) * DMODEL + t], a * Hsrc[(size_t)node * DMODEL + t]);
}

// pass 3: out = out / max(s,1e-9) + b
__global__ void gat_fin_kernel(float* __restrict__ out, const float* __restrict__ s,
                               const float* __restrict__ b, int sbase, int n)
{
    int row = blockIdx.x; if (row >= n) return;
    int t = threadIdx.x, h = t >> 5;
    float denom = fmaxf(s[(sbase + row) * NHEAD + h], 1e-9f);
    size_t i = (size_t)row * DMODEL + t;
    out[i] = out[i] / denom + b[t];
}

// fused residual + LayerNorm over D=256, one block per row
__global__ void add_ln_kernel(const float* __restrict__ x, const float* __restrict__ r,
                              const float* __restrict__ g, const float* __restrict__ b,
                              float* __restrict__ out, int n)
{
    __shared__ float red[8];
    int row = blockIdx.x; if (row >= n) return;
    int t = threadIdx.x;
    size_t base = (size_t)row * DMODEL;
    float v = x[base + t] + r[base + t];

    float partial = wave_sum32(v);
    if ((t & 31) == 0) red[t >> 5] = partial;
    __syncthreads();
    float mu = 0.f;
    #pragma unroll
    for (int i = 0; i < 8; i++) mu += red[i];
    mu *= (1.f / 256.f);
    __syncthreads();

    float w = v - mu;
    partial = wave_sum32(w * w);
    if ((t & 31) == 0) red[t >> 5] = partial;
    __syncthreads();
    float var = 0.f;
    #pragma unroll
    for (int i = 0; i < 8; i++) var += red[i];
    var *= (1.f / 256.f);

    out[base + t] = w * rsqrtf(var + 1e-5f) * g[t] + b[t];
}

// cross-attention over 3 memory tokens, one block per node, one wave per head
__global__ void cross_attn_kernel(const float* __restrict__ Q,
                                  const float* __restrict__ K0, const float* __restrict__ K1,
                                  const float* __restrict__ K2, const float* __restrict__ V0,
                                  const float* __restrict__ V1, const float* __restrict__ V2,
                                  float* __restrict__ out, int n)
{
    int row = blockIdx.x; if (row >= n) return;
    int t = threadIdx.x;
    size_t i = (size_t)row * DMODEL + t;
    float q  = Q[i];
    float k0 = K0[i], k1 = K1[i], k2 = K2[i];
    float v0 = V0[i], v1 = V1[i], v2 = V2[i];
    float d0 = wave_sum32(q * k0);
    float d1 = wave_sum32(q * k1);
    float d2 = wave_sum32(q * k2);
    const float scale = 0.1767766952966369f;   // 1/sqrt(32)
    d0 *= scale; d1 *= scale; d2 *= scale;
    float mx = fmaxf(d0, fmaxf(d1, d2));
    float e0 = expf(d0 - mx), e1 = expf(d1 - mx), e2 = expf(d2 - mx);
    float inv = 1.f / (e0 + e1 + e2);
    out[i] = (e0 * v0 + e1 * v1 + e2 * v2) * inv;
}

// ---------------------------------------------------------------------------
extern "C" void kernel_launch(void* const* d_in, const int* in_sizes, int n_in,
                              void* d_out, int out_size, void* d_ws, size_t ws_size,
                              hipStream_t stream)
{
    const float* Xp       = (const float*)d_in[0];
    const float* Xa       = (const float*)d_in[1];
    const float* W_pp     = (const float*)d_in[2];
    const float* al_pp    = (const float*)d_in[3];
    const float* ar_pp    = (const float*)d_in[4];
    const float* b_pp     = (const float*)d_in[5];
    const float* W_pa     = (const float*)d_in[6];
    const float* al_pa    = (const float*)d_in[7];
    const float* ar_pa    = (const float*)d_in[8];
    const float* b_pa     = (const float*)d_in[9];
    const float* sa_in_w  = (const float*)d_in[10];
    const float* sa_in_b  = (const float*)d_in[11];
    const float* sa_out_w = (const float*)d_in[12];
    const float* sa_out_b = (const float*)d_in[13];
    const float* ca_in_w  = (const float*)d_in[14];
    const float* ca_in_b  = (const float*)d_in[15];
    const float* ca_out_w = (const float*)d_in[16];
    const float* ca_out_b = (const float*)d_in[17];
    const float* ff1_w    = (const float*)d_in[18];
    const float* ff1_b    = (const float*)d_in[19];
    const float* ff2_w    = (const float*)d_in[20];
    const float* ff2_b    = (const float*)d_in[21];
    const float* ln1_g    = (const float*)d_in[22];
    const float* ln1_b    = (const float*)d_in[23];
    const float* ln2_g    = (const float*)d_in[24];
    const float* ln2_b    = (const float*)d_in[25];
    const float* ln3_g    = (const float*)d_in[26];
    const float* ln3_b    = (const float*)d_in[27];
    const int*   pp_src   = (const int*)d_in[28];
    const int*   pp_dst   = (const int*)d_in[29];
    const int*   ap_src   = (const int*)d_in[30];
    const int*   ap_dst   = (const int*)d_in[31];

    const int NP = NPOINT, NA = NAGENT, NT = NPOINT + NAGENT, D = DMODEL, E = NEDGE;
    const size_t U = (size_t)NP * D;         // 40000*256 floats

    float* w = (float*)d_ws;
    float* u[12];
    for (int i = 0; i < 12; i++) u[i] = w + (size_t)i * U;
    float* ffh   = w + 12 * U;               // 8192 x 2048 FFN chunk
    float* sm    = ffh + (size_t)8192 * DFFN;
    float* el_pp = sm;                 float* er_pp = el_pp + (size_t)NP * NHEAD;
    float* m_pp  = er_pp + (size_t)NP * NHEAD;   float* s_pp = m_pp + (size_t)NP * NHEAD;
    float* el_pa = s_pp + (size_t)NP * NHEAD;    float* er_pa = el_pa + (size_t)NT * NHEAD;
    float* m_pa  = er_pa + (size_t)NT * NHEAD;   float* s_pa = m_pa + (size_t)NT * NHEAD;

    auto gemm = [&](const float* A, const float* B, float* C, const float* bias,
                    int M, int N, int K, int opB, int relu) {
        dim3 grid((N + TN - 1) / TN, (M + TM - 1) / TM);
        gemm_wmma_f32<<<grid, dim3(256), 0, stream>>>(A, B, C, bias, M, N, K, opB, relu);
    };
    auto fill = [&](float* p, float v, size_t n) {
        fill_kernel<<<2048, 256, 0, stream>>>(p, v, n);
    };

    // ===================== GAT (point-point) =====================
    gemm(Xp, W_pp, u[0], nullptr, NP, D, D, 0, 0);                         // Hpp
    elr_kernel<<<NP, 256, 0, stream>>>(u[0], al_pp, ar_pp, el_pp, er_pp, NP);
    fill(m_pp, -INFINITY, (size_t)NP * NHEAD);
    fill(s_pp, 0.f, (size_t)NP * NHEAD);
    fill(u[1], 0.f, U);
    edge_max_kernel<<<(E * NHEAD + 255) / 256, 256, 0, stream>>>(pp_src, pp_dst, 0, el_pp, er_pp, m_pp, E);
    loop_max_kernel<<<(NP * NHEAD + 255) / 256, 256, 0, stream>>>(0, NP, el_pp, er_pp, m_pp);
    clamp_finite_kernel<<<256, 256, 0, stream>>>(m_pp, (size_t)NP * NHEAD);
    edge_acc_kernel<<<E, 256, 0, stream>>>(pp_src, pp_dst, 0, 0, el_pp, er_pp, m_pp, s_pp, u[1], u[0], E);
    loop_acc_kernel<<<NP, 256, 0, stream>>>(0, 0, el_pp, er_pp, m_pp, s_pp, u[1], u[0], NP);
    gat_fin_kernel<<<NP, 256, 0, stream>>>(u[1], s_pp, b_pp, 0, NP);       // h_pp in u1

    // ===================== GAT (agent-point) =====================
    gemm(Xa, W_pa, u[2], nullptr, NA, D, D, 0, 0);                         // Hpa rows [0,Na)
    gemm(Xp, W_pa, u[2] + (size_t)NA * D, nullptr, NP, D, D, 0, 0);        // Hpa rows [Na,Nt)
    elr_kernel<<<NT, 256, 0, stream>>>(u[2], al_pa, ar_pa, el_pa, er_pa, NT);
    fill(m_pa, -INFINITY, (size_t)NT * NHEAD);
    fill(s_pa, 0.f, (size_t)NT * NHEAD);
    fill(u[4], 0.f, U);
    edge_max_kernel<<<(E * NHEAD + 255) / 256, 256, 0, stream>>>(ap_src, ap_dst, NA, el_pa, er_pa, m_pa, E);
    loop_max_kernel<<<(NP * NHEAD + 255) / 256, 256, 0, stream>>>(NA, NP, el_pa, er_pa, m_pa);
    clamp_finite_kernel<<<256, 256, 0, stream>>>(m_pa, (size_t)NT * NHEAD);
    edge_acc_kernel<<<E, 256, 0, stream>>>(ap_src, ap_dst, NA, NA, el_pa, er_pa, m_pa, s_pa, u[4], u[2], E);
    loop_acc_kernel<<<NP, 256, 0, stream>>>(NA, NA, el_pa, er_pa, m_pa, s_pa, u[4], u[2], NP);
    gat_fin_kernel<<<NP, 256, 0, stream>>>(u[4], s_pa, b_pa, NA, NP);      // h_ap in u4

    // ===================== self-attention (seq len 1 => out = V-proj) =====================
    gemm(Xp, sa_in_w + 2 * D * D, u[6], sa_in_b + 2 * D, NP, D, D, 1, 0);  // V = Xp@wv^T+bv
    gemm(u[6], sa_out_w, u[7], sa_out_b, NP, D, D, 1, 0);                  // out proj
    add_ln_kernel<<<NP, 256, 0, stream>>>(Xp, u[7], ln1_g, ln1_b, u[5], NP); // x1 in u5

    // ===================== cross-attention (mem = [Xp, h_pp, h_ap]) =====================
    gemm(u[5], ca_in_w,             u[0], ca_in_b,         NP, D, D, 1, 0);  // Q
    gemm(Xp,   ca_in_w + D * D,     u[2], ca_in_b + D,     NP, D, D, 1, 0);  // K0
    gemm(u[1], ca_in_w + D * D,     u[3], ca_in_b + D,     NP, D, D, 1, 0);  // K1
    gemm(u[4], ca_in_w + D * D,     u[6], ca_in_b + D,     NP, D, D, 1, 0);  // K2
    gemm(Xp,   ca_in_w + 2 * D * D, u[7], ca_in_b + 2 * D, NP, D, D, 1, 0);  // V0
    gemm(u[1], ca_in_w + 2 * D * D, u[8], ca_in_b + 2 * D, NP, D, D, 1, 0);  // V1
    gemm(u[4], ca_in_w + 2 * D * D, u[9], ca_in_b + 2 * D, NP, D, D, 1, 0);  // V2
    cross_attn_kernel<<<NP, 256, 0, stream>>>(u[0], u[2], u[3], u[6], u[7], u[8], u[9], u[10], NP);
    gemm(u[10], ca_out_w, u[2], ca_out_b, NP, D, D, 1, 0);                 // attn out proj
    add_ln_kernel<<<NP, 256, 0, stream>>>(u[5], u[2], ln2_g, ln2_b, u[11], NP); // x2 in u11

    // ===================== FFN (chunked rows to bound scratch) =====================
    for (int c = 0; c < NP; c += 8192) {
        int Mc = NP - c < 8192 ? NP - c : 8192;
        gemm(u[11] + (size_t)c * D, ff1_w, ffh, ff1_b, Mc, DFFN, D, 1, 1);       // relu(x2@W1^T+b1)
        gemm(ffh, ff2_w, u[10] + (size_t)c * D, ff2_b, Mc, D, DFFN, 1, 0);       // @W2^T+b2
    }
    add_ln_kernel<<<NP, 256, 0, stream>>>(u[11], u[10], ln3_g, ln3_b, (float*)d_out, NP);
}